// MotorHead_43293270344325
// MI455X (gfx1250) — compile-verified
//
#include <hip/hip_runtime.h>

typedef __attribute__((ext_vector_type(2))) float v2f;
typedef __attribute__((ext_vector_type(8))) float v8f;

#define B_    4
#define L_    1024
#define V_    1280
#define NE_   768
#define NBIN_ 8
#define NHID_ 32
#define NOUT1 (NBIN_ * NHID_)   // 256
#define CHUNK 32
#define NCHUNK (L_ / CHUNK)     // 32

// ---------------------------------------------------------------------------
// GEMM1: x[M=4096, N=256] = h[4096, 768] @ W1[768, 256]   (f32 WMMA 16x16x4)
// One wave computes a 32x64 tile (2 M-tiles x 4 N-tiles = 8 accumulators);
// B operands are shared across the two M-tiles: 8 WMMAs per K-step for
// 2 b64 A-loads + 8 b32 B-loads.
// ---------------------------------------------------------------------------
__global__ __launch_bounds__(128) void gemm1_kernel(const float* __restrict__ h,
                                                    const float* __restrict__ W1,
                                                    float* __restrict__ x) {
  const int wave  = blockIdx.x * 4 + (threadIdx.x >> 5);
  const int lane  = threadIdx.x & 31;
  const int mTile = wave >> 2;            // 4096/32 = 128 M-tiles
  const int nTile = wave & 3;             // 256/64  = 4 N-groups
  const int mRow  = mTile * 32;
  const int nCol  = nTile * 64;
  const int hi    = lane >> 4;            // 0: K pair {0,1}, 1: K pair {2,3}
  const int lo    = lane & 15;

  v8f p0 = {}, p1 = {}, p2 = {}, p3 = {};   // rows mRow..mRow+15
  v8f q0 = {}, q1 = {}, q2 = {}, q3 = {};   // rows mRow+16..mRow+31
  const float* Arow0 = h + (size_t)(mRow + lo) * NE_;
  const float* Arow1 = Arow0 + (size_t)16 * NE_;

  for (int k = 0; k < NE_; k += 4) {
    const int ka = k + (hi << 1);
    v2f a0, a1;
    a0.x = Arow0[ka];  a0.y = Arow0[ka + 1];
    a1.x = Arow1[ka];  a1.y = Arow1[ka + 1];
    const float* Bp = W1 + (size_t)ka * NOUT1 + nCol + lo;
    v2f b0, b1, b2, b3;
    b0.x = Bp[0];   b0.y = Bp[NOUT1];
    b1.x = Bp[16];  b1.y = Bp[NOUT1 + 16];
    b2.x = Bp[32];  b2.y = Bp[NOUT1 + 32];
    b3.x = Bp[48];  b3.y = Bp[NOUT1 + 48];
    p0 = __builtin_amdgcn_wmma_f32_16x16x4_f32(false, a0, false, b0, (short)0, p0, false, false);
    p1 = __builtin_amdgcn_wmma_f32_16x16x4_f32(false, a0, false, b1, (short)0, p1, false, false);
    p2 = __builtin_amdgcn_wmma_f32_16x16x4_f32(false, a0, false, b2, (short)0, p2, false, false);
    p3 = __builtin_amdgcn_wmma_f32_16x16x4_f32(false, a0, false, b3, (short)0, p3, false, false);
    q0 = __builtin_amdgcn_wmma_f32_16x16x4_f32(false, a1, false, b0, (short)0, q0, false, false);
    q1 = __builtin_amdgcn_wmma_f32_16x16x4_f32(false, a1, false, b1, (short)0, q1, false, false);
    q2 = __builtin_amdgcn_wmma_f32_16x16x4_f32(false, a1, false, b2, (short)0, q2, false, false);
    q3 = __builtin_amdgcn_wmma_f32_16x16x4_f32(false, a1, false, b3, (short)0, q3, false, false);
  }

  const int rOff = hi ? 8 : 0;
  float* D0 = x + (size_t)(mRow + rOff) * NOUT1 + nCol + lo;
  float* D1 = D0 + (size_t)16 * NOUT1;
#pragma unroll
  for (int i = 0; i < 8; ++i) {
    D0[(size_t)i * NOUT1 + 0]  = p0[i];
    D0[(size_t)i * NOUT1 + 16] = p1[i];
    D0[(size_t)i * NOUT1 + 32] = p2[i];
    D0[(size_t)i * NOUT1 + 48] = p3[i];
    D1[(size_t)i * NOUT1 + 0]  = q0[i];
    D1[(size_t)i * NOUT1 + 16] = q1[i];
    D1[(size_t)i * NOUT1 + 32] = q2[i];
    D1[(size_t)i * NOUT1 + 48] = q3[i];
  }
}

// ---------------------------------------------------------------------------
// GEMM2: logits[M=32768, N=1280] = x[32768, 32] @ W2[32, 1280]
// Row m = (b*L + l)*8 + n maps exactly onto x flat layout (lda = 32).
// Same 32x64 wave tile with B reuse across M-tiles.
// ---------------------------------------------------------------------------
__global__ __launch_bounds__(128) void gemm2_kernel(const float* __restrict__ x,
                                                    const float* __restrict__ W2,
                                                    float* __restrict__ logits) {
  const int wave  = blockIdx.x * 4 + (threadIdx.x >> 5);
  const int lane  = threadIdx.x & 31;
  const int nGroups = V_ / 64;            // 20
  const int mTile = wave / nGroups;       // 32768/32 = 1024 M-tiles
  const int nTile = wave % nGroups;
  const int mRow  = mTile * 32;
  const int nCol  = nTile * 64;
  const int hi    = lane >> 4;
  const int lo    = lane & 15;

  v8f p0 = {}, p1 = {}, p2 = {}, p3 = {};
  v8f q0 = {}, q1 = {}, q2 = {}, q3 = {};
  const float* Arow0 = x + (size_t)(mRow + lo) * NHID_;
  const float* Arow1 = Arow0 + (size_t)16 * NHID_;

#pragma unroll
  for (int k = 0; k < NHID_; k += 4) {
    const int ka = k + (hi << 1);
    v2f a0, a1;
    a0.x = Arow0[ka];  a0.y = Arow0[ka + 1];
    a1.x = Arow1[ka];  a1.y = Arow1[ka + 1];
    const float* Bp = W2 + (size_t)ka * V_ + nCol + lo;
    v2f b0, b1, b2, b3;
    b0.x = Bp[0];   b0.y = Bp[V_];
    b1.x = Bp[16];  b1.y = Bp[V_ + 16];
    b2.x = Bp[32];  b2.y = Bp[V_ + 32];
    b3.x = Bp[48];  b3.y = Bp[V_ + 48];
    p0 = __builtin_amdgcn_wmma_f32_16x16x4_f32(false, a0, false, b0, (short)0, p0, false, false);
    p1 = __builtin_amdgcn_wmma_f32_16x16x4_f32(false, a0, false, b1, (short)0, p1, false, false);
    p2 = __builtin_amdgcn_wmma_f32_16x16x4_f32(false, a0, false, b2, (short)0, p2, false, false);
    p3 = __builtin_amdgcn_wmma_f32_16x16x4_f32(false, a0, false, b3, (short)0, p3, false, false);
    q0 = __builtin_amdgcn_wmma_f32_16x16x4_f32(false, a1, false, b0, (short)0, q0, false, false);
    q1 = __builtin_amdgcn_wmma_f32_16x16x4_f32(false, a1, false, b1, (short)0, q1, false, false);
    q2 = __builtin_amdgcn_wmma_f32_16x16x4_f32(false, a1, false, b2, (short)0, q2, false, false);
    q3 = __builtin_amdgcn_wmma_f32_16x16x4_f32(false, a1, false, b3, (short)0, q3, false, false);
  }

  const int rOff = hi ? 8 : 0;
  float* D0 = logits + (size_t)(mRow + rOff) * V_ + nCol + lo;
  float* D1 = D0 + (size_t)16 * V_;
#pragma unroll
  for (int i = 0; i < 8; ++i) {
    D0[(size_t)i * V_ + 0]  = p0[i];
    D0[(size_t)i * V_ + 16] = p1[i];
    D0[(size_t)i * V_ + 32] = p2[i];
    D0[(size_t)i * V_ + 48] = p3[i];
    D1[(size_t)i * V_ + 0]  = q0[i];
    D1[(size_t)i * V_ + 16] = q1[i];
    D1[(size_t)i * V_ + 32] = q2[i];
    D1[(size_t)i * V_ + 48] = q3[i];
  }
}

// ---------------------------------------------------------------------------
// C1: per (b, chunk) first-occurrence table: table[b,c,v] = min{ j in chunk :
//     targets[b,j]==v } (global index), else L.
// ---------------------------------------------------------------------------
__global__ __launch_bounds__(256) void tte_table_kernel(const int* __restrict__ targets,
                                                        int* __restrict__ table) {
  const int b = blockIdx.y;
  const int c = blockIdx.x;
  __shared__ int tgt[CHUNK];
  if (threadIdx.x < CHUNK)
    tgt[threadIdx.x] = targets[b * L_ + c * CHUNK + threadIdx.x];
  __syncthreads();
  for (int v = threadIdx.x; v < V_; v += 256) {
    int idx = L_;
    for (int j = CHUNK - 1; j >= 0; --j)
      if (tgt[j] == v) idx = c * CHUNK + j;
    table[((size_t)b * NCHUNK + c) * V_ + v] = idx;
  }
}

// ---------------------------------------------------------------------------
// C2: per (b, chunk): seed last[v] from later-chunk tables, walk rows
// backwards updating last[targets[r]] = r; emit tte and all 8 censor bins.
// Reference quirk: token_index[b, r>0, 0] is forced to 0 (below-diagonal
// targets_exp zeros scatter index 0 with src j<r, min j = 0).
// ---------------------------------------------------------------------------
__global__ __launch_bounds__(256) void tte_mask_kernel(const float* __restrict__ age,
                                                       const float* __restrict__ targets_age,
                                                       const int* __restrict__ targets,
                                                       const int* __restrict__ table,
                                                       float* __restrict__ tte_out,
                                                       float* __restrict__ mask_out) {
  const int b = blockIdx.y;
  const int c = blockIdx.x;
  __shared__ int last[V_];
  __shared__ int tgt[CHUNK];

  for (int v = threadIdx.x; v < V_; v += 256) {
    int m = L_;
    for (int c2 = c + 1; c2 < NCHUNK; ++c2) {
      const int t = table[((size_t)b * NCHUNK + c2) * V_ + v];
      m = t < m ? t : m;
    }
    last[v] = m;
  }
  if (threadIdx.x < CHUNK)
    tgt[threadIdx.x] = targets[b * L_ + c * CHUNK + threadIdx.x];
  __syncthreads();

  const float* ta = targets_age + (size_t)b * L_;
  for (int j = CHUNK - 1; j >= 0; --j) {
    const int r = c * CHUNK + j;
    if (threadIdx.x == 0) last[tgt[j]] = r;   // r < any previously stored index
    __syncthreads();
    const float ageval = age[(size_t)b * L_ + r];
    const size_t row = (size_t)b * L_ + r;
    float* ttep  = tte_out  + row * V_;
    float* maskp = mask_out + row * (size_t)(NBIN_ * V_);
    for (int v = threadIdx.x; v < V_; v += 256) {
      int raw = last[v];
      if (v == 0 && r > 0) raw = 0;
      const bool noev = (raw == L_);
      const int  ti   = noev ? (L_ - 1) : raw;
      const float t   = ta[ti] - ageval;
      ttep[v] = t;
#pragma unroll
      for (int n = 0; n < NBIN_; ++n) {
        const float lo = 1.25f * (float)n;     // edges exactly representable
        const float hi = lo + 1.25f;
        const bool m = ((t >= lo) && (t < hi)) || noev;
        maskp[(size_t)n * V_ + v] = m ? 1.0f : 0.0f;
      }
    }
    __syncthreads();
  }
}

// ---------------------------------------------------------------------------
extern "C" void kernel_launch(void* const* d_in, const int* in_sizes, int n_in,
                              void* d_out, int out_size, void* d_ws, size_t ws_size,
                              hipStream_t stream) {
  (void)in_sizes; (void)n_in; (void)out_size; (void)ws_size;
  const float* h           = (const float*)d_in[0];
  const float* age         = (const float*)d_in[1];
  const float* targets_age = (const float*)d_in[2];
  // d_in[3] = delta_t (unused by the outputs)
  const int*   targets     = (const int*)d_in[4];
  const float* W1          = (const float*)d_in[5];
  const float* W2          = (const float*)d_in[6];

  float* out    = (float*)d_out;
  float* logits = out;                                             // B*L*8*1280
  float* tte    = logits + (size_t)B_ * L_ * NBIN_ * V_;           // B*L*1280
  float* mask   = tte    + (size_t)B_ * L_ * V_;                   // B*L*8*1280

  float* x     = (float*)d_ws;                                     // 4096*256 f32
  int*   table = (int*)(x + (size_t)B_ * L_ * NOUT1);              // B*NCHUNK*V

  // GEMM1: (4096/32)*(256/64) = 512 waves, 4 waves/block -> 128 blocks
  gemm1_kernel<<<128, 128, 0, stream>>>(h, W1, x);

  // occurrence table (independent of GEMMs)
  dim3 gScan(NCHUNK, B_);
  tte_table_kernel<<<gScan, 256, 0, stream>>>(targets, table);

  // GEMM2: (32768/32)*(1280/64) = 20480 waves, 4 waves/block -> 5120 blocks
  gemm2_kernel<<<5120, 128, 0, stream>>>(x, W2, logits);

  // fused tte + censor_mask pass
  tte_mask_kernel<<<gScan, 256, 0, stream>>>(age, targets_age, targets, table,
                                             tte, mask);
}